// Decoder_429496730360
// MI455X (gfx1250) — compile-verified
//
#include <hip/hip_runtime.h>

#define BB 32
#define SS 2048
#define HH 1024
#define K3H 3072
#define SPLIT 8

typedef float v2f __attribute__((ext_vector_type(2)));
typedef float v4f __attribute__((ext_vector_type(4)));
typedef float v8f __attribute__((ext_vector_type(8)));

__device__ __forceinline__ float wave_sum(float v) {
#pragma unroll
  for (int o = 16; o > 0; o >>= 1) v += __shfl_xor(v, o, 32);
  return v;
}

// Non-temporal b128 load: encoder_out is a 536MB single-use stream (2.8x the
// 192MB L2) -> TH=NT so it doesn't thrash lines that have actual reuse.
__device__ __forceinline__ v4f ld_nt4(const float* p) {
  return __builtin_nontemporal_load((const v4f*)p);
}
__device__ __forceinline__ v2f ld_nt2(const float* p) {
  return __builtin_nontemporal_load((const v2f*)p);
}

// ---------------------------------------------------------------------------
// Kernel 1: fused scores + online-softmax + weighted context, single pass over
// encoder_out (536 MB read exactly once, NT-hinted). Grid = B*SPLIT blocks,
// 512 threads. Each wave keeps a 2048-float context accumulator (64 VGPRs per
// lane) and online (m, l); block partials written to workspace.
// ---------------------------------------------------------------------------
__global__ __launch_bounds__(512) void attn_partial_kernel(
    const float* __restrict__ x, const float* __restrict__ eo,
    const float* __restrict__ Wenergy, const float* __restrict__ benergy,
    float* __restrict__ ctx_part, float* __restrict__ m_part,
    float* __restrict__ l_part) {
  __shared__ float ctx_sh[2 * HH];
  __shared__ float m_sh[16];
  __shared__ float l_tot;

  const int tid = threadIdx.x;
  const int w = tid >> 5;
  const int lane = tid & 31;
  const int b = blockIdx.x >> 3;   // batch
  const int p = blockIdx.x & 7;    // S-split part

  for (int i = tid; i < 2 * HH; i += 512) ctx_sh[i] = 0.f;
  if (tid == 0) l_tot = 0.f;

  // preload "we" (2048 floats) into registers, coalesced float4 layout
  const v4f* we4 = (const v4f*)(Wenergy + HH);
  v4f wreg[16];
#pragma unroll
  for (int j = 0; j < 16; ++j) wreg[j] = we4[j * 32 + lane];

  // xq[b] . wq  (wave-redundant, cheap)
  const v4f* xq4 = (const v4f*)(x + (size_t)b * HH);
  const v4f* wq4 = (const v4f*)(Wenergy);
  float xw = 0.f;
#pragma unroll
  for (int j = 0; j < 8; ++j) {
    v4f t = xq4[j * 32 + lane] * wq4[j * 32 + lane];
    xw += t.x + t.y + t.z + t.w;
  }
  xw = wave_sum(xw);
  const float bias = xw + benergy[0];

  float m = -3.0e38f, l = 0.f;
  v4f acc[16];
#pragma unroll
  for (int j = 0; j < 16; ++j) acc[j] = (v4f)0.f;

  const int s_base = p * (SS / SPLIT);
  for (int i = 0; i < (SS / SPLIT) / 16; ++i) {
    const int s = s_base + w + i * 16;
    const float* row = eo + ((size_t)b * SS + s) * (2 * HH);
    v4f r[16];
#pragma unroll
    for (int j = 0; j < 16; ++j)
      r[j] = ld_nt4(row + (j * 32 + lane) * 4);  // 16x b128 NT, coalesced

    float d = 0.f;
#pragma unroll
    for (int j = 0; j < 16; ++j) {
      v4f t = r[j] * wreg[j];
      d += t.x + t.y + t.z + t.w;
    }
    d = wave_sum(d);
    const float score = fmaxf(d + bias, 0.f);  // relu

    if (score > m) {  // wave-uniform branch; rare after warm-up
      const float c = __expf(m - score);
      l = l * c + 1.f;
#pragma unroll
      for (int j = 0; j < 16; ++j) acc[j] = acc[j] * c + r[j];
      m = score;
    } else {
      const float pr = __expf(score - m);
      l += pr;
#pragma unroll
      for (int j = 0; j < 16; ++j) acc[j] += pr * r[j];
    }
  }

  // combine the 16 waves of this block
  if (lane == 0) m_sh[w] = m;
  __syncthreads();
  float M = m_sh[0];
#pragma unroll
  for (int j = 1; j < 16; ++j) M = fmaxf(M, m_sh[j]);
  const float scale = __expf(m - M);
  if (lane == 0) atomicAdd(&l_tot, l * scale);
#pragma unroll
  for (int j = 0; j < 16; ++j) {
    const int base = (j * 32 + lane) * 4;
    atomicAdd(&ctx_sh[base + 0], acc[j].x * scale);
    atomicAdd(&ctx_sh[base + 1], acc[j].y * scale);
    atomicAdd(&ctx_sh[base + 2], acc[j].z * scale);
    atomicAdd(&ctx_sh[base + 3], acc[j].w * scale);
  }
  __syncthreads();

  float* cp = ctx_part + (size_t)blockIdx.x * (2 * HH);
  for (int i = tid; i < 2 * HH; i += 512) cp[i] = ctx_sh[i];
  if (tid == 0) {
    m_part[blockIdx.x] = M;
    l_part[blockIdx.x] = l_tot;
  }
}

// ---------------------------------------------------------------------------
// Kernel 2: combine SPLIT partials per batch -> normalized context, and build
// gru_in = [xq | context] (32 x 3072) in workspace.
// ---------------------------------------------------------------------------
__global__ __launch_bounds__(256) void attn_combine_kernel(
    const float* __restrict__ x, const float* __restrict__ ctx_part,
    const float* __restrict__ m_part, const float* __restrict__ l_part,
    float* __restrict__ gru) {
  const int b = blockIdx.x;
  const int tid = threadIdx.x;

  float M = -3.0e38f;
#pragma unroll
  for (int p = 0; p < SPLIT; ++p) M = fmaxf(M, m_part[b * SPLIT + p]);
  float sc[SPLIT];
  float lt = 0.f;
#pragma unroll
  for (int p = 0; p < SPLIT; ++p) {
    sc[p] = __expf(m_part[b * SPLIT + p] - M);
    lt += sc[p] * l_part[b * SPLIT + p];
  }
  const float inv = 1.f / lt;

  for (int d = tid; d < 2 * HH; d += 256) {
    float s = 0.f;
#pragma unroll
    for (int p = 0; p < SPLIT; ++p)
      s += sc[p] * ctx_part[(size_t)(b * SPLIT + p) * (2 * HH) + d];
    gru[(size_t)b * K3H + HH + d] = s * inv;
  }
  for (int d = tid; d < HH; d += 256)
    gru[(size_t)b * K3H + d] = x[(size_t)b * HH + d];
}

// ---------------------------------------------------------------------------
// Kernel 3: gi = gru_in @ w_ih^T + b_ih via V_WMMA_F32_16X16X4_F32, fused with
// gate activations and the double store of h. Block owns 16 output columns
// across all 3 gates (r/z/n); 4 waves split K=3072, ds_add_f32 merge in LDS.
// w_ih (37.7MB) is a single-use stream -> NT loads.
// ---------------------------------------------------------------------------
__global__ __launch_bounds__(128) void gru_wmma_kernel(
    const float* __restrict__ gru, const float* __restrict__ w_ih,
    const float* __restrict__ b_ih, const float* __restrict__ b_hh,
    float* __restrict__ out) {
  __shared__ float gi_sh[3][512];  // [gate][m*16 + c], m=0..31, c=0..15

  const int tid = threadIdx.x;
  const int w = tid >> 5;
  const int lane = tid & 31;
  const int t16 = blockIdx.x * 16;

  // init with b_ih so the WMMA partial sums land on top of the bias
  for (int i = tid; i < 512; i += 128) {
    const int c = i & 15;
    gi_sh[0][i] = b_ih[t16 + c];
    gi_sh[1][i] = b_ih[HH + t16 + c];
    gi_sh[2][i] = b_ih[2 * HH + t16 + c];
  }
  __syncthreads();

  // A (16x4 f32): VGPR0 = K{0|2}, VGPR1 = K{1|3} split at lane 16 -> per-lane
  // contiguous float2 at column offset koff. B (4x16) mirrors A's K split.
  const int kbase = w * (K3H / 4);  // each wave: 768-wide K slice
  const int koff = (lane >> 4) * 2;
  const int ml = lane & 15;
  const float* pA0 = gru + (size_t)ml * K3H + kbase + koff;
  const float* pA1 = gru + (size_t)(16 + ml) * K3H + kbase + koff;
  const float* pBr = w_ih + (size_t)(t16 + ml) * K3H + kbase + koff;
  const float* pBz = w_ih + (size_t)(HH + t16 + ml) * K3H + kbase + koff;
  const float* pBn = w_ih + (size_t)(2 * HH + t16 + ml) * K3H + kbase + koff;

  v8f dR0 = {}, dR1 = {}, dZ0 = {}, dZ1 = {}, dN0 = {}, dN1 = {};
  for (int kk = 0; kk < 192; ++kk) {
    const int off = kk * 4;
    v2f a0 = *(const v2f*)(pA0 + off);
    v2f a1 = *(const v2f*)(pA1 + off);
    v2f br = ld_nt2(pBr + off);
    v2f bz = ld_nt2(pBz + off);
    v2f bn = ld_nt2(pBn + off);
    dR0 = __builtin_amdgcn_wmma_f32_16x16x4_f32(false, a0, false, br, (short)0, dR0, false, false);
    dR1 = __builtin_amdgcn_wmma_f32_16x16x4_f32(false, a1, false, br, (short)0, dR1, false, false);
    dZ0 = __builtin_amdgcn_wmma_f32_16x16x4_f32(false, a0, false, bz, (short)0, dZ0, false, false);
    dZ1 = __builtin_amdgcn_wmma_f32_16x16x4_f32(false, a1, false, bz, (short)0, dZ1, false, false);
    dN0 = __builtin_amdgcn_wmma_f32_16x16x4_f32(false, a0, false, bn, (short)0, dN0, false, false);
    dN1 = __builtin_amdgcn_wmma_f32_16x16x4_f32(false, a1, false, bn, (short)0, dN1, false, false);
  }

  // D layout: VGPR v, lanes<16 -> M=v, lanes>=16 -> M=v+8; N = lane&15
  const int mhi = (lane >> 4) * 8;
#pragma unroll
  for (int v = 0; v < 8; ++v) {
    const int m0 = v + mhi;
    const int m1 = 16 + v + mhi;
    atomicAdd(&gi_sh[0][m0 * 16 + ml], dR0[v]);
    atomicAdd(&gi_sh[0][m1 * 16 + ml], dR1[v]);
    atomicAdd(&gi_sh[1][m0 * 16 + ml], dZ0[v]);
    atomicAdd(&gi_sh[1][m1 * 16 + ml], dZ1[v]);
    atomicAdd(&gi_sh[2][m0 * 16 + ml], dN0[v]);
    atomicAdd(&gi_sh[2][m1 * 16 + ml], dN1[v]);
  }
  __syncthreads();

  // gates + double store (out and hn are the same 32x1024 block of h)
  for (int i = tid; i < 512; i += 128) {
    const int m = i >> 4;
    const int c = i & 15;
    const int ig = t16 + c;
    const float r = 1.f / (1.f + __expf(-(gi_sh[0][i] + b_hh[ig])));
    const float z = 1.f / (1.f + __expf(-(gi_sh[1][i] + b_hh[HH + ig])));
    const float n = tanhf(gi_sh[2][i] + r * b_hh[2 * HH + ig]);
    const float h = (1.f - z) * n;
    out[(size_t)m * HH + ig] = h;
    out[(size_t)BB * HH + (size_t)m * HH + ig] = h;
  }
}

extern "C" void kernel_launch(void* const* d_in, const int* in_sizes, int n_in,
                              void* d_out, int out_size, void* d_ws,
                              size_t ws_size, hipStream_t stream) {
  (void)in_sizes; (void)n_in; (void)out_size; (void)ws_size;
  const float* x = (const float*)d_in[0];        // (32,1,1024)
  const float* eo = (const float*)d_in[1];       // (32,2048,2048)
  const float* We = (const float*)d_in[3];       // (1,3072)
  const float* be = (const float*)d_in[4];       // (1,)
  const float* w_ih = (const float*)d_in[5];     // (3072,3072)
  const float* b_ih = (const float*)d_in[7];     // (3072,)
  const float* b_hh = (const float*)d_in[8];     // (3072,)
  // d_in[2] (encoder_hn) and d_in[6] (w_hh) are unused by the reference.

  float* ws = (float*)d_ws;
  float* ctx_part = ws;                   // 32*8*2048 = 524288 floats
  float* m_part = ws + 524288;            // 256 floats
  float* l_part = ws + 524544;            // 256 floats
  float* gru = ws + 524800;               // 32*3072  =  98304 floats

  attn_partial_kernel<<<BB * SPLIT, 512, 0, stream>>>(x, eo, We, be, ctx_part,
                                                      m_part, l_part);
  attn_combine_kernel<<<BB, 256, 0, stream>>>(x, ctx_part, m_part, l_part, gru);
  gru_wmma_kernel<<<HH / 16, 128, 0, stream>>>(gru, w_ih, b_ih, b_hh,
                                               (float*)d_out);
}